// SS2D_66511863546364
// MI455X (gfx1250) — compile-verified
//
#include <hip/hip_runtime.h>
#include <math.h>

// ---------------- problem constants ----------------
constexpr int B_ = 8, D_ = 192, H_ = 64, W_ = 64;
constexpr int L_ = H_ * W_;          // 4096
constexpr int K_ = 4, N_ = 16, R_ = 6;
constexpr int C_ = R_ + 2 * N_;      // 38
constexpr int C48 = 48;              // padded row count (3 WMMA M-tiles)

typedef __attribute__((ext_vector_type(2))) float v2f;
typedef __attribute__((ext_vector_type(8))) float v8f;

// CrossScan index permutation: offset within the H*W plane for direction k,
// sequence position l.
//  k=0: row-major  l
//  k=1: col-major  (w*H+h scan) -> plane offset h*W+w  == 6-bit swap of l
//  k=2: reversed row-major
//  k=3: reversed col-major
__device__ __forceinline__ int xs_index(int k, int l) {
    int p = (k & 2) ? (L_ - 1 - l) : l;
    if (k & 1) p = ((p & 63) << 6) | (p >> 6);
    return p;
}

__device__ __forceinline__ float softplus_f(float v) {
    return (v > 20.0f) ? v : __logf(1.0f + __expf(v));
}

// ---------------------------------------------------------------------------
// Kernel 1: fused CrossScan + x_proj einsum ('bkdl,kcd->bkcl') via
// V_WMMA_F32_16X16X4_F32.  Out stored transposed: xdbl_t[b,k,l,c48].
// Per (b,k): Out[48 x 4096] = Wpad[48 x 192] * XS[192 x 4096]
// Block = 256 thr (8 waves); each block: one (b,k), 128 l-columns.
// Wave w owns N-tile (16 cols) w; runs all 3 M-tiles, K loop of 48 wmma steps.
// A frag (16x4 f32): lanes 0-15 M rows, VGPR0/1 = K 0/1; lanes 16-31 = K 2/3.
// B frag (4x16 f32): lanes 0-15 N cols, VGPR0/1 = K 0/1; lanes 16-31 = K 2/3.
// ---------------------------------------------------------------------------
__global__ __launch_bounds__(256) void proj_wmma(
    const float* __restrict__ x,       // (B,D,H,W)
    const float* __restrict__ xw,      // (K,38,192)
    float* __restrict__ xdbl_t)        // (B,K,L,48)
{
    __shared__ float Wlds[C48 * D_];   // 36 KB, rows >= 38 zero-padded

    const int ltiles = L_ / 128;
    int ltile = blockIdx.x % ltiles;
    int bk    = blockIdx.x / ltiles;
    int k = bk % K_;
    int b = bk / K_;

    for (int i = threadIdx.x; i < C48 * D_; i += 256) {
        int c = i / D_, d = i - c * D_;
        Wlds[i] = (c < C_) ? xw[((size_t)k * C_ + c) * D_ + d] : 0.0f;
    }
    __syncthreads();

    const int wave   = threadIdx.x >> 5;
    const int lane   = threadIdx.x & 31;
    const int lane15 = lane & 15;
    const int half   = lane >> 4;          // K-half for A/B fragments

    const int l0   = ltile * 128 + wave * 16;
    const int lcol = l0 + lane15;          // this lane's N column
    const int poff = xs_index(k, lcol);    // CrossScan gather offset (const over kb)

    const float* xb = x + (size_t)b * D_ * L_;

    v8f acc0 = {}, acc1 = {}, acc2 = {};

    for (int kb = 0; kb < D_ / 4; ++kb) {
        int d0 = kb * 4 + 2 * half;
        v2f bf;
        bf[0] = xb[(size_t)d0 * L_ + poff];
        bf[1] = xb[(size_t)(d0 + 1) * L_ + poff];

        v2f a0, a1, a2;
        a0[0] = Wlds[(0 * 16 + lane15) * D_ + d0];
        a0[1] = Wlds[(0 * 16 + lane15) * D_ + d0 + 1];
        a1[0] = Wlds[(1 * 16 + lane15) * D_ + d0];
        a1[1] = Wlds[(1 * 16 + lane15) * D_ + d0 + 1];
        a2[0] = Wlds[(2 * 16 + lane15) * D_ + d0];
        a2[1] = Wlds[(2 * 16 + lane15) * D_ + d0 + 1];

        acc0 = __builtin_amdgcn_wmma_f32_16x16x4_f32(false, a0, false, bf,
                                                     (short)0, acc0, false, false);
        acc1 = __builtin_amdgcn_wmma_f32_16x16x4_f32(false, a1, false, bf,
                                                     (short)0, acc1, false, false);
        acc2 = __builtin_amdgcn_wmma_f32_16x16x4_f32(false, a2, false, bf,
                                                     (short)0, acc2, false, false);
    }

    // C/D layout: N = lane15 (column l), M = v + 8*half (row c). Store
    // transposed (l-major): lane writes 8 contiguous floats per tile.
    float* outp = xdbl_t + ((size_t)(b * K_ + k) * L_ + l0 + lane15) * C48;
#pragma unroll
    for (int v = 0; v < 8; ++v) {
        outp[0 * 16 + 8 * half + v] = acc0[v];
        outp[1 * 16 + 8 * half + v] = acc1[v];
        outp[2 * 16 + 8 * half + v] = acc2[v];
    }
}

// ---------------------------------------------------------------------------
// Kernel 2: selective scan. One wave = two (b,k,d) channels; lane n = state
// dim. dt-projection (R=6 dot) + softplus computed inline from the contiguous
// 48-float xdbl column. ys stored (b,k,l,d) so merge reads coalesce over d.
// ---------------------------------------------------------------------------
__global__ __launch_bounds__(256) void scan_kernel(
    const float* __restrict__ x,        // (B,D,L)
    const float* __restrict__ xdbl_t,   // (B,K,L,48)
    const float* __restrict__ dtw,      // (K,D,R)
    const float* __restrict__ dtb,      // (K,D)
    const float* __restrict__ A_logs,   // (K*D, N)
    const float* __restrict__ Ds,       // (K*D)
    float* __restrict__ ys)             // (B,K,L,D)
{
    const int wave = threadIdx.x >> 5;
    const int lane = threadIdx.x & 31;
    const int n    = lane & 15;
    const int half = lane >> 4;

    const int dtiles = D_ / 16;
    int dt = blockIdx.x % dtiles;
    int bk = blockIdx.x / dtiles;
    int k  = bk % K_;
    int b  = bk / K_;
    int d  = dt * 16 + wave * 2 + half;
    int kd = k * D_ + d;

    const float Aln   = -__expf(A_logs[(size_t)kd * N_ + n]);
    const float Dskip = Ds[kd];
    const float bias  = dtb[kd];
    float w0 = dtw[(size_t)kd * R_ + 0], w1 = dtw[(size_t)kd * R_ + 1];
    float w2 = dtw[(size_t)kd * R_ + 2], w3 = dtw[(size_t)kd * R_ + 3];
    float w4 = dtw[(size_t)kd * R_ + 4], w5 = dtw[(size_t)kd * R_ + 5];

    const float* xrow = x + ((size_t)b * D_ + d) * L_;
    const float* colb = xdbl_t + (size_t)(b * K_ + k) * L_ * C48;
    float* yout       = ys + (size_t)(b * K_ + k) * L_ * D_ + d;

    float h = 0.0f;
    for (int l = 0; l < L_; ++l) {
        const float* col = colb + (size_t)l * C48;
        __builtin_prefetch(col + 8 * C48, 0, 1);   // global_prefetch_b8

        float4 c03 = *(const float4*)col;
        float2 c45 = *(const float2*)(col + 4);
        float sraw = bias + w0 * c03.x + w1 * c03.y + w2 * c03.z +
                            w3 * c03.w + w4 * c45.x + w5 * c45.y;
        float delta = softplus_f(sraw);

        float u  = xrow[xs_index(k, l)];
        float Bn = col[R_ + n];
        float Cn = col[R_ + N_ + n];

        h = __expf(delta * Aln) * h + (delta * u) * Bn;

        float y = h * Cn;                 // reduce over 16-lane state group
        y += __shfl_xor(y, 1, 32);
        y += __shfl_xor(y, 2, 32);
        y += __shfl_xor(y, 4, 32);
        y += __shfl_xor(y, 8, 32);

        if (n == 0) yout[(size_t)l * D_] = y + Dskip * u;
    }
}

// ---------------------------------------------------------------------------
// Kernel 3: CrossMerge + LayerNorm over channels. One block per (b,l),
// 192 threads = one channel each (6 waves).
// ---------------------------------------------------------------------------
__global__ __launch_bounds__(192) void merge_ln(
    const float* __restrict__ ys,       // (B,K,L,D)
    const float* __restrict__ ln_w,
    const float* __restrict__ ln_b,
    float* __restrict__ out)            // (B,H,W,D)
{
    __shared__ float red_s[6], red_s2[6];

    int bl = blockIdx.x;
    int b  = bl / L_;
    int l  = bl % L_;
    int d  = threadIdx.x;

    int lT = ((l & 63) << 6) | (l >> 6);   // (h,w) position in the WH scan

    const float* base = ys + (size_t)b * K_ * L_ * D_;
    float v = base[((size_t)0 * L_ + l) * D_ + d]
            + base[((size_t)2 * L_ + (L_ - 1 - l)) * D_ + d]
            + base[((size_t)1 * L_ + lT) * D_ + d]
            + base[((size_t)3 * L_ + (L_ - 1 - lT)) * D_ + d];

    float s = v, s2 = v * v;
    s  += __shfl_xor(s, 1, 32);  s2 += __shfl_xor(s2, 1, 32);
    s  += __shfl_xor(s, 2, 32);  s2 += __shfl_xor(s2, 2, 32);
    s  += __shfl_xor(s, 4, 32);  s2 += __shfl_xor(s2, 4, 32);
    s  += __shfl_xor(s, 8, 32);  s2 += __shfl_xor(s2, 8, 32);
    s  += __shfl_xor(s, 16, 32); s2 += __shfl_xor(s2, 16, 32);

    int wv = d >> 5;
    if ((d & 31) == 0) { red_s[wv] = s; red_s2[wv] = s2; }
    __syncthreads();

    float ts = 0.0f, ts2 = 0.0f;
#pragma unroll
    for (int i = 0; i < 6; ++i) { ts += red_s[i]; ts2 += red_s2[i]; }

    float mu   = ts * (1.0f / D_);
    float var  = ts2 * (1.0f / D_) - mu * mu;
    float rstd = rsqrtf(var + 1e-5f);

    out[(size_t)bl * D_ + d] = (v - mu) * rstd * ln_w[d] + ln_b[d];
}

// ---------------------------------------------------------------------------
extern "C" void kernel_launch(void* const* d_in, const int* in_sizes, int n_in,
                              void* d_out, int out_size, void* d_ws, size_t ws_size,
                              hipStream_t stream) {
    const float* x     = (const float*)d_in[0];   // (8,192,64,64)
    const float* xw    = (const float*)d_in[1];   // (4,38,192)
    const float* dtw   = (const float*)d_in[2];   // (4,192,6)
    const float* dtb   = (const float*)d_in[3];   // (4,192)
    const float* Alog  = (const float*)d_in[4];   // (768,16)
    const float* Dsk   = (const float*)d_in[5];   // (768)
    const float* lnw   = (const float*)d_in[6];
    const float* lnb   = (const float*)d_in[7];
    float* out = (float*)d_out;

    float* xdbl_t = (float*)d_ws;                              // 25.2 MB
    float* ysbuf  = xdbl_t + (size_t)B_ * K_ * L_ * C48;       // 100.7 MB

    proj_wmma<<<B_ * K_ * (L_ / 128), 256, 0, stream>>>(x, xw, xdbl_t);
    scan_kernel<<<B_ * K_ * (D_ / 16), 256, 0, stream>>>(x, xdbl_t, dtw, dtb,
                                                         Alog, Dsk, ysbuf);
    merge_ln<<<B_ * L_, 192, 0, stream>>>(ysbuf, lnw, lnb, out);
}